// FFM_Layer_83537113907381
// MI455X (gfx1250) — compile-verified
//
#include <hip/hip_runtime.h>

// ---------------------------------------------------------------------------
// FFM layer for MI455X (gfx1250, wave32).
//   out[b] = w0 + dense[b,:]@w[:13] + sum_s w[idx(b,s)]
//          + 0.5*( || sum_f field_f[b,f,:] ||^2  -  || field_f[b] ||_F^2 )
//   field_f[b] = dense[b,:] @ v[:13]  (GEMM -> v_wmma_f32_16x16x4_f32)
//              + sum_s v[idx(b,s)]    (bandwidth-bound gathers, b128 coalesced)
// ---------------------------------------------------------------------------

#define DENSE_NUM   13
#define SPARSE_NUM  26
#define FEAT_NUM    10000
#define FIELD_NUM   39
#define KDIM        8
#define SLOTS       (FIELD_NUM * KDIM)   // 312 floats per embedding row
#define NPAD        320                  // 312 padded to 20 tiles of 16
#define NTILES      (NPAD / 16)          // 20
#define ROWS_PER_BLOCK 16
#define THREADS     256                  // 8 wave32s

typedef float v2f __attribute__((ext_vector_type(2)));
typedef float v8f __attribute__((ext_vector_type(8)));

__global__ __launch_bounds__(THREADS)
void ffm_layer_kernel(const float* __restrict__ dense,   // [B,13]
                      const int*   __restrict__ sparse,  // [B,26]
                      const float* __restrict__ w0,      // [1]
                      const float* __restrict__ w,       // [260013]
                      const float* __restrict__ v,       // [260013, 312]
                      float* __restrict__ out,           // [B]
                      int batch)
{
    __shared__ __align__(16) float sDense[ROWS_PER_BLOCK][16]; // zero-padded A
    __shared__ __align__(16) float sB[NPAD][16];               // v[:16,:320]^T, zero-padded
    __shared__ __align__(16) float sFF[ROWS_PER_BLOCK][NPAD];  // field_f accum

    const int tid   = threadIdx.x;
    const int lane  = tid & 31;
    // force into SGPR: wave index is uniform per wave -> scalar loops below
    const int waveS = __builtin_amdgcn_readfirstlane(tid >> 5); // 0..7
    const int rowBase = blockIdx.x * ROWS_PER_BLOCK;

    // ---- stage dense rows (zero padded) ----
    if (tid < ROWS_PER_BLOCK * 16) {
        const int r = tid >> 4, c = tid & 15;
        float val = 0.0f;
        if (c < DENSE_NUM && (rowBase + r) < batch)
            val = dense[(rowBase + r) * DENSE_NUM + c];
        sDense[r][c] = val;
    }
    // ---- stage B panel transposed: sB[col][k] = v[k, col], pads = 0 ----
    for (int i = tid; i < NPAD * 16; i += THREADS) {
        const int k = i / NPAD;          // 0..15  (feature d)
        const int c = i % NPAD;          // 0..319 (flat field*K slot)
        float val = 0.0f;
        if (k < DENSE_NUM && c < SLOTS)
            val = v[(long)k * SLOTS + c];
        sB[c][k] = val;
    }
    __syncthreads();

    // ================= Phase A: dense einsum via V_WMMA_F32_16X16X4_F32 =====
    // A frag (16x4 f32): lanes 0-15 hold K={0,1}, lanes 16-31 hold K={2,3}.
    // B frag (4x16 f32): same K split, N = lane & 15.
    // C/D (16x16 f32):   VGPR r -> M = r + 8*half, N = lane & 15.
    const int m    = lane & 15;
    const int half = lane >> 4;

    v2f aFrag[4];
#pragma unroll
    for (int kc = 0; kc < 4; ++kc) {
        const int k0 = kc * 4 + half * 2;                  // even -> 8B aligned
        aFrag[kc] = *(const v2f*)&sDense[m][k0];
    }

    for (int tile = waveS; tile < NTILES; tile += 8) {     // scalar loop
        v8f acc = {};
        const int col = tile * 16 + m;
#pragma unroll
        for (int kc = 0; kc < 4; ++kc) {
            const int k0 = kc * 4 + half * 2;
            const v2f b = *(const v2f*)&sB[col][k0];       // ds_load_b64
            acc = __builtin_amdgcn_wmma_f32_16x16x4_f32(
                      false, aFrag[kc], false, b,
                      (short)0, acc, false, false);
        }
#pragma unroll
        for (int r = 0; r < 8; ++r)
            sFF[r + half * 8][tile * 16 + m] = acc[r];     // pads get exact 0
    }
    __syncthreads();

    // ============ Phases B+C: per-wave ownership of 2 batch rows ============
    for (int rr = 0; rr < 2; ++rr) {
        const int row  = waveS * 2 + rr;                   // scalar
        const int brow = rowBase + row;
        if (brow >= batch) continue;                       // scalar branch

        // lanes 0..25 each hold one global gather index for this row
        int myGidx = 0;
        if (lane < SPARSE_NUM)
            myGidx = sparse[brow * SPARSE_NUM + lane] + DENSE_NUM + lane * FEAT_NUM;

        // ---- Phase B: 26 gathers of 312 floats, coalesced b128 loads ----
        float4 a0 = make_float4(0.f, 0.f, 0.f, 0.f);
        float4 a1 = a0, a2 = a0;
        for (int s = 0; s < SPARSE_NUM; ++s) {
            // SGPR row index -> saddr-form global_load_b128
            const int g = __builtin_amdgcn_readlane(myGidx, s);
            const float4* b4 = (const float4*)(v + (long)g * SLOTS); // 16B aligned
            float4 x0 = b4[lane];                 // slots   0..127
            float4 x1 = b4[32 + lane];            // slots 128..255
            a0.x += x0.x; a0.y += x0.y; a0.z += x0.z; a0.w += x0.w;
            a1.x += x1.x; a1.y += x1.y; a1.z += x1.z; a1.w += x1.w;
            if (lane < 14) {                      // slots 256..311 (56 floats)
                float4 x2 = b4[64 + lane];
                a2.x += x2.x; a2.y += x2.y; a2.z += x2.z; a2.w += x2.w;
            }
        }
        {   // merge into LDS (wave owns this row exclusively)
            float4* ff = (float4*)&sFF[row][0];
            float4 c0 = ff[lane];
            c0.x += a0.x; c0.y += a0.y; c0.z += a0.z; c0.w += a0.w;
            ff[lane] = c0;
            float4 c1 = ff[32 + lane];
            c1.x += a1.x; c1.y += a1.y; c1.z += a1.z; c1.w += a1.w;
            ff[32 + lane] = c1;
            if (lane < 14) {
                float4 c2 = ff[64 + lane];
                c2.x += a2.x; c2.y += a2.y; c2.z += a2.z; c2.w += a2.w;
                ff[64 + lane] = c2;
            }
        }

        // ---- Phase C: second-order + first-order reduction ----
        // slot = lane + 32*r  ->  k = slot % 8 = lane % 8 for all r
        float ssq = 0.0f, sk = 0.0f;
#pragma unroll
        for (int r = 0; r < NPAD / 32; ++r) {     // 10; slots 312..319 are 0
            const float x = sFF[row][lane + 32 * r];
            ssq += x * x;
            sk  += x;
        }
        // reduce sk within each k-group (lanes sharing lane%8)
        sk += __shfl_xor(sk, 8, 32);
        sk += __shfl_xor(sk, 16, 32);
        float z = sk * sk;                        // each k replicated in 4 lanes
#pragma unroll
        for (int d = 1; d < 32; d <<= 1) {
            z   += __shfl_xor(z,   d, 32);
            ssq += __shfl_xor(ssq, d, 32);
        }
        const float second = 0.5f * (z * 0.25f - ssq);

        // first order: lanes 0..12 dense*w, lanes 0..25 gathered w
        float t = 0.0f;
        if (lane < DENSE_NUM)  t += sDense[row][lane] * w[lane];
        if (lane < SPARSE_NUM) t += w[myGidx];
#pragma unroll
        for (int d = 1; d < 32; d <<= 1) t += __shfl_xor(t, d, 32);

        if (lane == 0) out[brow] = w0[0] + t + second;
    }
}

extern "C" void kernel_launch(void* const* d_in, const int* in_sizes, int n_in,
                              void* d_out, int out_size, void* d_ws, size_t ws_size,
                              hipStream_t stream) {
    const float* dense  = (const float*)d_in[0];   // [B,13] f32
    const int*   sparse = (const int*)  d_in[1];   // [B,26] i32
    const float* w0     = (const float*)d_in[2];   // [1]    f32
    const float* w      = (const float*)d_in[3];   // [260013,1] f32
    const float* v      = (const float*)d_in[4];   // [260013,39,8] f32
    float*       out    = (float*)d_out;           // [B,1]  f32

    const int batch  = in_sizes[0] / DENSE_NUM;    // 4096
    const int blocks = (batch + ROWS_PER_BLOCK - 1) / ROWS_PER_BLOCK;

    ffm_layer_kernel<<<blocks, THREADS, 0, stream>>>(dense, sparse, w0, w, v,
                                                     out, batch);
}